// Smooth_35287451304942
// MI455X (gfx1250) — compile-verified
//
#include <hip/hip_runtime.h>

typedef __attribute__((ext_vector_type(2))) float v2f;
typedef __attribute__((ext_vector_type(8))) float v8f;

#define T_DIM   4096
#define C_DIM   64
#define TILES_T 256                 // T_DIM / 16
#define TILES_C 4                   // C_DIM / 16
#define TGROUP  4                   // time-tiles chained per wave
#define WAVES_PER_BLOCK 8

// ---------------- compile-time Savitzky-Golay weights -----------------------
// Cubic least-squares projection on symmetric points u,v in [-3,3].
// Normal-equation moments S0=7,S2=28,S4=196,S6=1588 decouple even/odd:
//   even block inv = 1/588  * [[196,-28],[-28,7]]
//   odd  block inv = 1/6048 * [[1588,-196],[-196,28]]
constexpr float proj_c(float u, float v) {
    float u2 = u * u, v2 = v * v;
    return (196.0f - 28.0f * (u2 + v2) + 7.0f * u2 * v2) / 588.0f
         + u * v * (1588.0f - 196.0f * (u2 + v2) + 28.0f * u2 * v2) / 6048.0f;
}

// W[r, j] of the 16x24 per-tile matrix (j = input time t0-3+j, r = output row).
// cls: 0 interior, 1 first time-tile (head rows), 2 last time-tile (tail rows).
constexpr float wElem(int cls, int r, int j) {
    float u = 0.0f;
    int center = r + 3;                                  // interior FIR: Proj(0,d) = (7-d^2)/21
    if (cls == 1 && r < 3)        { u = (float)(r - 3);  center = 6;  }   // P_head
    else if (cls == 2 && r >= 13) { u = (float)(r - 12); center = 15; }   // P_tail
    int d = j - center;
    return (d < -3 || d > 3) ? 0.0f : proj_c(u, (float)d);
}

// Table in WMMA A-operand layout: entry [cls][q][lane] = (W[lo, 4q+2hi], W[lo, 4q+2hi+1])
// with lo = lane&15 (= M), hi = lane>>4 (K half). One v2f load per chunk, fully coalesced.
struct alignas(16) WTab { float w[3][6][32][2]; };

constexpr WTab makeWTab() {
    WTab t{};
    for (int cls = 0; cls < 3; ++cls)
        for (int q = 0; q < 6; ++q)
            for (int lane = 0; lane < 32; ++lane) {
                int hi = lane >> 4, lo = lane & 15;
                t.w[cls][q][lane][0] = wElem(cls, lo, 4 * q + 2 * hi);
                t.w[cls][q][lane][1] = wElem(cls, lo, 4 * q + 2 * hi + 1);
            }
    return t;
}

__constant__ WTab WTAB = makeWTab();

// ---------------------------------- kernel ----------------------------------
// One wave computes TGROUP=4 consecutive 16(time)x16(channel) output tiles of
// one batch row:  Y(16x16) = W(16x24) x X(24x16) via 6 K-chunks of
// V_WMMA_F32_16X16X4_F32.  Tile i's B chunks 4,5 (rows t0+13..t0+20, identical
// lane mapping) are reused as tile i+1's chunks 0,1  ->  read amp 72/64.
__global__ void __launch_bounds__(WAVES_PER_BLOCK * 32)
savgol_wmma_kernel(const float* __restrict__ x, float* __restrict__ y) {
    const int lane = threadIdx.x & 31;
    const int lo   = lane & 15;          // N (channel) for B/D, M (row) for A
    const int hi   = lane >> 4;          // half-wave: K half (A/B), M half (D)

    const int g  = blockIdx.x * WAVES_PER_BLOCK + (threadIdx.x >> 5);
    const int gc = g & (TILES_C - 1);            // channel tile
    const int gt = (g >> 2) & (TILES_T / TGROUP - 1);   // time group (0..63)
    const int b  = g >> 8;                       // batch (4*64 groups per batch)

    const int  t00 = gt << 6;                    // first time of group (64 rows)
    const int  c0  = gc << 4;
    const bool edgeGroup = (gt == 0) || (gt == TILES_T / TGROUP - 1);

    const float* __restrict__ xb = x + (size_t)b * (T_DIM * C_DIM) + c0 + lo;
    float*       __restrict__ yb = y + (size_t)b * (T_DIM * C_DIM) + c0 + lo;

    auto loadrow = [&](int t) -> float {
        if (edgeGroup) t = t < 0 ? 0 : (t > T_DIM - 1 ? T_DIM - 1 : t);  // OOB rows have W==0
        return xb[t * C_DIM];
    };

    // Interior weights; edge groups overwrite per tile below.
    v2f wreg[6];
    #pragma unroll
    for (int q = 0; q < 6; ++q)
        wreg[q] = *(const v2f*)&WTAB.w[0][q][lane][0];

    v2f bv[6];

    #pragma unroll
    for (int i = 0; i < TGROUP; ++i) {
        const int t0 = t00 + (i << 4);

        if (i == 0) {
            #pragma unroll
            for (int q = 0; q < 6; ++q) {
                const int ra = t0 - 3 + 4 * q + 2 * hi;
                bv[q].x = loadrow(ra);
                bv[q].y = loadrow(ra + 1);
            }
        } else {
            bv[0] = bv[4];               // register chaining: rows t0+13..t0+16
            bv[1] = bv[5];               //                    rows t0+17..t0+20
            #pragma unroll
            for (int q = 2; q < 6; ++q) {
                const int ra = t0 - 3 + 4 * q + 2 * hi;
                bv[q].x = loadrow(ra);
                bv[q].y = loadrow(ra + 1);
            }
        }

        if (edgeGroup) {                 // wave-uniform; rare (2 of 64 groups)
            const int tt  = (t00 >> 4) + i;
            const int cls = (tt == 0) ? 1 : ((tt == TILES_T - 1) ? 2 : 0);
            #pragma unroll
            for (int q = 0; q < 6; ++q)
                wreg[q] = *(const v2f*)&WTAB.w[cls][q][lane][0];
        }

        v8f acc = {};
        #pragma unroll
        for (int q = 0; q < 6; ++q) {
            // D(16x16) += A(16x4) * B(4x16), all f32
            acc = __builtin_amdgcn_wmma_f32_16x16x4_f32(
                /*neg_a=*/false, wreg[q], /*neg_b=*/false, bv[q],
                /*c_mod=*/(short)0, acc, /*reuse_a=*/false, /*reuse_b=*/false);
        }

        // D layout: VGPR v -> row M = v + 8*hi, col N = lo.  Output is never
        // re-read: non-temporal stores keep the input halo resident in L2.
        #pragma unroll
        for (int v = 0; v < 8; ++v)
            __builtin_nontemporal_store(acc[v], yb + (t0 + v + 8 * hi) * C_DIM);
    }
}

extern "C" void kernel_launch(void* const* d_in, const int* in_sizes, int n_in,
                              void* d_out, int out_size, void* d_ws, size_t ws_size,
                              hipStream_t stream) {
    const float* x = (const float*)d_in[0];
    float* y = (float*)d_out;

    const int B = in_sizes[0] / (T_DIM * C_DIM);                    // 256
    const int groups = B * (TILES_T / TGROUP) * TILES_C;            // 65536 waves
    const int blocks = groups / WAVES_PER_BLOCK;                    // 8192

    savgol_wmma_kernel<<<dim3(blocks), dim3(WAVES_PER_BLOCK * 32), 0, stream>>>(x, y);
}